// RealNDimFourier_41987600285961
// MI455X (gfx1250) — compile-verified
//
#include <hip/hip_runtime.h>
#include <math.h>

typedef float v2f __attribute__((ext_vector_type(2)));
typedef float v8f __attribute__((ext_vector_type(8)));

#define HH 512
#define WW 512
#define PI2 6.28318530717958647692f

// One block = one (plane, 16-row tile-strip). 8 waves/block, each wave does
// four 16x16 WMMA output tiles. A (16x4) and B (512x4) factors live in LDS.
__global__ __launch_bounds__(256) void RealNDimFourier_kernel(
    const float* __restrict__ x,
    const int*   __restrict__ idx_h,
    const int*   __restrict__ idx_w,
    const float* __restrict__ phase_u,
    const float* __restrict__ strengths,
    float*       __restrict__ out)
{
    __shared__ float sA[16][4];    // y-factor for this tile strip
    __shared__ float sB[512][4];   // x-factor (K and strength folded in)

    const int p  = blockIdx.y;     // plane = b*4 + c
    const int tr = blockIdx.x;     // tile-row within plane
    const int y0 = tr << 4;
    const int b  = p >> 2;
    const int c  = p & 3;
    const int k  = b + 48;         // last-write-wins scatter entry for this plane

    const int   h0  = idx_h[k];
    const int   w0  = idx_w[k];
    const float phi = PI2 * phase_u[k];
    const int   h1  = 511 - h0;
    const int   w1  = 511 - w0;

    // analytic ||g||^2 via Fourier orthogonality (1/N of ifft cancels in normalize)
    const float d0 = (((h0 & 255) == 0) && ((w0 & 255) == 0)) ? 1.0f : 0.0f;
    const float d1 = (((h1 & 255) == 0) && ((w1 & 255) == 0)) ? 1.0f : 0.0f;
    const float N  = (float)(HH * WW);
    const float norm2 = 0.5f * N * ((2.0f + d0 + d1) + phi * phi * (2.0f - d0 - d1));
    const bool  valid = (c == (b & 3));           // plane actually received a scatter
    const float K = valid ? (16.0f * strengths[p] * rsqrtf(norm2))
                          : __builtin_nanf("");   // empty plane -> 0/0 -> NaN (as reference)

    const int t = threadIdx.x;

    // B factor: exact angle reduction  theta = 2*pi*((w*x) mod 512)/512
    for (int xx = t; xx < 512; xx += 256) {
        const int m0 = (w0 * xx) & 511;
        const int m1 = (w1 * xx) & 511;
        const float a0 = (float)m0 * (PI2 / 512.0f);
        const float a1 = (float)m1 * (PI2 / 512.0f);
        const float c0 = __cosf(a0), s0 = __sinf(a0);
        const float c1 = __cosf(a1), s1 = __sinf(a1);
        sB[xx][0] = K * ( c0 - phi * s0);
        sB[xx][1] = K * (-s0 - phi * c0);
        sB[xx][2] = K * ( c1 - phi * s1);
        sB[xx][3] = K * (-s1 - phi * c1);
    }
    if (t < 16) {
        const int y  = y0 + t;
        const int m0 = (h0 * y) & 511;
        const int m1 = (h1 * y) & 511;
        const float a0 = (float)m0 * (PI2 / 512.0f);
        const float a1 = (float)m1 * (PI2 / 512.0f);
        sA[t][0] = __cosf(a0);
        sA[t][1] = __sinf(a0);
        sA[t][2] = __cosf(a1);
        sA[t][3] = __sinf(a1);
    }
    __syncthreads();

    const int lane = t & 31;       // wave32
    const int wv   = t >> 5;       // 8 waves
    const int n    = lane & 15;
    const int half = lane >> 4;

    // A operand (16x4 f32, 2 VGPRs): lane L holds A[L%16][2*(L/16)+{0,1}]
    v2f a;
    a.x = sA[n][2 * half + 0];
    a.y = sA[n][2 * half + 1];

    const int planeBase = (p * (HH * WW)) + y0 * WW;

    #pragma unroll
    for (int tci = 0; tci < 4; ++tci) {
        const int x0 = ((wv << 2) + tci) << 4;

        // B operand (4x16 f32): symmetric striping, bank-conflict-free ds_load_b64
        v2f bb;
        bb.x = sB[x0 + n][2 * half + 0];
        bb.y = sB[x0 + n][2 * half + 1];

        v8f acc = {};
        acc = __builtin_amdgcn_wmma_f32_16x16x4_f32(
            false, a, false, bb, (short)0, acc, false, false);

        // D layout: vgpr r -> row y0 + r + 8*half, col x0 + n
        const int base = planeBase + (half * 8) * WW + x0 + n;
        #pragma unroll
        for (int r = 0; r < 8; ++r) {
            const int off = base + r * WW;
            float v = x[off] + acc[r];
            // branchy clamp so NaN propagates exactly like jnp.clip
            v = (v > 1.0f) ? 1.0f : ((v < 0.0f) ? 0.0f : v);
            out[off] = v;
        }
    }
}

extern "C" void kernel_launch(void* const* d_in, const int* in_sizes, int n_in,
                              void* d_out, int out_size, void* d_ws, size_t ws_size,
                              hipStream_t stream) {
    const float* x         = (const float*)d_in[0];
    const int*   idx_h     = (const int*)d_in[1];
    const int*   idx_w     = (const int*)d_in[2];
    const float* phase_u   = (const float*)d_in[3];
    const float* strengths = (const float*)d_in[4];
    float*       out       = (float*)d_out;

    dim3 grid(32, 64);   // 32 tile-rows x 64 planes
    RealNDimFourier_kernel<<<grid, 256, 0, stream>>>(x, idx_h, idx_w, phase_u, strengths, out);
}